// DGCNNModule_65000035058613
// MI455X (gfx1250) — compile-verified
//
#include <hip/hip_runtime.h>

#define BB   16
#define NN   4096
#define CIN  3
#define COUT 64
#define KK   20
#define NPTS  (BB*NN)        // 65536
#define NEDGE (NPTS*KK)      // 1310720
#define FLT_BIG 3.402823466e+38f

typedef float v2f __attribute__((ext_vector_type(2)));
typedef float v8f __attribute__((ext_vector_type(8)));

// ---------------- Kernel Z: zero the stats accumulators ----------------
__global__ void k_zero(float* stats) {
  if (threadIdx.x < 32) stats[threadIdx.x] = 0.f;
}

// ---------------- Kernel A: KNN (top-20) + edge second-moment stats ----------------
__global__ __launch_bounds__(256) void k_knn(const float* __restrict__ x,
                                             int* __restrict__ idxws,
                                             float* __restrict__ stats) {
  __shared__ float pts[NN * CIN];                       // 48 KB
  const int b    = blockIdx.x >> 4;
  const int tile = blockIdx.x & 15;
  const int tid  = threadIdx.x;
  for (int i = tid; i < NN * CIN; i += 256) pts[i] = x[b * NN * CIN + i];
  __syncthreads();

  const int q = tile * 256 + tid;
  const float qx = pts[q*3+0], qy = pts[q*3+1], qz = pts[q*3+2];

  float bd[KK]; int bi[KK];
  #pragma unroll
  for (int j = 0; j < KK; ++j) { bd[j] = FLT_BIG; bi[j] = 0; }
  float worst = FLT_BIG; int wslot = 0;

  for (int m = 0; m < NN; ++m) {
    float dx = pts[m*3+0] - qx;
    float dy = pts[m*3+1] - qy;
    float dz = pts[m*3+2] - qz;
    float d = dx*dx + dy*dy + dz*dz;
    if (d < worst) {                 // strict < keeps earliest index on ties
      #pragma unroll
      for (int j = 0; j < KK; ++j) if (j == wslot) { bd[j] = d; bi[j] = m; }
      float nw = -FLT_BIG; int ns = 0;
      #pragma unroll
      for (int j = 0; j < KK; ++j) if (bd[j] > nw) { nw = bd[j]; ns = j; }
      worst = nw; wslot = ns;
    }
  }

  // edge moments: e = [xi(3), xj-xi(3)] ; acc[0..5]=sum(e), acc[6..26]=sum upper-tri e*e
  float acc[27];
  #pragma unroll
  for (int i = 0; i < 27; ++i) acc[i] = 0.f;
  #pragma unroll
  for (int j = 0; j < KK; ++j) {
    int m = bi[j];
    float e[6];
    e[0] = qx; e[1] = qy; e[2] = qz;
    e[3] = pts[m*3+0] - qx; e[4] = pts[m*3+1] - qy; e[5] = pts[m*3+2] - qz;
    #pragma unroll
    for (int r = 0; r < 6; ++r) acc[r] += e[r];
    int t = 6;
    #pragma unroll
    for (int r = 0; r < 6; ++r)
      #pragma unroll
      for (int s = r; s < 6; ++s) { acc[t] += e[r] * e[s]; ++t; }
  }

  const int p = b * NN + q;
  #pragma unroll
  for (int j = 0; j < KK; ++j) idxws[p * KK + j] = b * NN + bi[j];  // global row idx

  #pragma unroll
  for (int i = 0; i < 27; ++i) {
    float v = acc[i];
    for (int off = 16; off >= 1; off >>= 1) v += __shfl_xor(v, off);
    if ((threadIdx.x & 31) == 0) atomicAdd(&stats[i], v);
  }
}

// ---------------- Kernel B: fold moments into per-channel BN scale/shift ----------------
__global__ void k_bnstats(const float* __restrict__ stats,
                          const float* __restrict__ w1,
                          const float* __restrict__ b1,
                          const float* __restrict__ gamma,
                          const float* __restrict__ beta,
                          float* __restrict__ sbuf,
                          float* __restrict__ tbuf) {
  int c = threadIdx.x;
  if (c >= COUT) return;
  const float inv = 1.0f / (float)NEDGE;
  float mu[6], M[6][6];
  for (int r = 0; r < 6; ++r) mu[r] = stats[r] * inv;
  int t = 6;
  for (int r = 0; r < 6; ++r)
    for (int s = r; s < 6; ++s) { float v = stats[t++] * inv; M[r][s] = v; M[s][r] = v; }
  float w[6];
  for (int r = 0; r < 6; ++r) w[r] = w1[r * COUT + c];
  float dwm = 0.f;
  for (int r = 0; r < 6; ++r) dwm += w[r] * mu[r];
  float b = b1[c];
  float mean = dwm + b;
  float exx = 0.f;
  for (int r = 0; r < 6; ++r)
    for (int s = 0; s < 6; ++s) exx += w[r] * M[r][s] * w[s];
  float eh2 = exx + 2.f * b * dwm + b * b;
  float var = eh2 - mean * mean;
  float sc  = gamma[c] * rsqrtf(var + 1e-5f);
  // h_bn = D*sc + tc, with D = e@w1 (bias folded in)
  sbuf[c] = sc;
  tbuf[c] = (b - mean) * sc + beta[c];
}

// ---------------- Kernel C: fused gather -> WMMA matmul1 -> BN/ReLU -> WMMA matmul2 -> max_k ----------------
#define WPB 8          // waves per block
#define PPW 8          // points per wave
#define STR 66         // padded LDS row stride (even -> 8B-aligned b64 A loads)

__global__ __launch_bounds__(256) void k_edgeconv(
    const float* __restrict__ x,
    const int*   __restrict__ idxws,
    const float* __restrict__ w1,
    const float* __restrict__ w2,
    const float* __restrict__ b2,
    const float* __restrict__ sbuf,
    const float* __restrict__ tbuf,
    float* __restrict__ out) {
  __shared__ __align__(16) v2f  w2p[32 * COUT];          // row-pairs of w2, 16 KB
  __shared__ __align__(16) float stage[WPB][16 * STR];   // 33 KB

  const int tid = threadIdx.x;
  for (int i = tid; i < 32 * COUT; i += 256) {
    int pr = i >> 6, cc = i & 63;
    v2f tmp;
    tmp.x = w2[(2 * pr + 0) * COUT + cc];
    tmp.y = w2[(2 * pr + 1) * COUT + cc];
    w2p[pr * COUT + cc] = tmp;
  }
  __syncthreads();

  const int wv   = tid >> 5;
  const int lane = tid & 31;
  const int half = lane >> 4;    // 0: lanes 0-15, 1: lanes 16-31
  const int l16  = lane & 15;
  const int koff = half * 2;     // A/B K offset inside a k-step
  float* st = &stage[wv][0];

  // ---- loop-invariant per-lane operands (loaded once) ----
  v2f  w1b0[4], w1b1[4];
  float scv[4], tcv[4], b2v[4];
  #pragma unroll
  for (int t = 0; t < 4; ++t) {
    const int col = t * 16 + l16;
    w1b0[t].x = w1[(koff + 0) * COUT + col];
    w1b0[t].y = w1[(koff + 1) * COUT + col];
    w1b1[t].x = half ? 0.f : w1[4 * COUT + col];  // K rows 6,7 are zero pad
    w1b1[t].y = half ? 0.f : w1[5 * COUT + col];
    scv[t] = sbuf[col];
    tcv[t] = tbuf[col];
    b2v[t] = b2[col];
  }

  for (int pi = 0; pi < PPW; ++pi) {
    const int p = (blockIdx.x * WPB + wv) * PPW + pi;
    const float xi0 = x[p*3+0], xi1 = x[p*3+1], xi2 = x[p*3+2];

    float rmax[4];
    #pragma unroll
    for (int t = 0; t < 4; ++t) rmax[t] = -FLT_BIG;

    #pragma unroll
    for (int mt = 0; mt < 2; ++mt) {         // M-tiles: edges 0-15, 16-19(+pad)
      // ---- build edge-feature row M = l16 (branch-free: mask by validity) ----
      const int ed = mt * 16 + l16;
      const float vm = (ed < KK) ? 1.f : 0.f;
      const int edc  = (ed < KK) ? ed : (KK - 1);
      const int nb = idxws[p * KK + edc];
      const float xj0 = x[nb*3+0], xj1 = x[nb*3+1], xj2 = x[nb*3+2];
      const float e0 = xi0 * vm, e1 = xi1 * vm, e2 = xi2 * vm;
      const float e3 = (xj0 - xi0) * vm, e4 = (xj1 - xi1) * vm, e5 = (xj2 - xi2) * vm;

      v2f a0, a1;                            // A-layout: lanes<16 hold K=koff..koff+1
      a0.x = half ? e2  : e0;  a0.y = half ? e3  : e1;
      a1.x = half ? 0.f : e4;  a1.y = half ? 0.f : e5;   // K=6,7 padded to 0

      // ---- matmul1: acc1 = e(16x8pad) @ w1(8pad x 64), two k-steps ----
      v8f acc1[4];
      #pragma unroll
      for (int t = 0; t < 4; ++t)
        #pragma unroll
        for (int j = 0; j < 8; ++j) acc1[t][j] = 0.f;

      #pragma unroll
      for (int t = 0; t < 4; ++t) {
        acc1[t] = __builtin_amdgcn_wmma_f32_16x16x4_f32(false, a0, false, w1b0[t], (short)0, acc1[t], false, false);
        acc1[t] = __builtin_amdgcn_wmma_f32_16x16x4_f32(false, a1, false, w1b1[t], (short)0, acc1[t], false, false);
      }

      // ---- BN + ReLU in C-layout, stage 16x64 tile to LDS row-major ----
      #pragma unroll
      for (int t = 0; t < 4; ++t) {
        const int col = t * 16 + l16;
        #pragma unroll
        for (int j = 0; j < 8; ++j) {
          float v = acc1[t][j] * scv[t] + tcv[t];
          v = v > 0.f ? v : 0.f;
          st[(j + half * 8) * STR + col] = v;
        }
      }
      asm volatile("s_wait_dscnt 0" ::: "memory");       // stores visible wave-wide

      // ---- matmul2: acc2 = h1r(16x64) @ w2(64x64) + b2, 16 k-steps ----
      v8f acc2[4];
      #pragma unroll
      for (int t = 0; t < 4; ++t)
        #pragma unroll
        for (int j = 0; j < 8; ++j) acc2[t][j] = b2v[t];

      #pragma unroll
      for (int kk = 0; kk < 16; ++kk) {
        const int c0 = 4 * kk + koff;                    // even -> 8B aligned
        const v2f a = *(const v2f*)&st[l16 * STR + c0];  // ds_load_b64
        const int pr = 2 * kk + half;                    // row-pair index
        #pragma unroll
        for (int t = 0; t < 4; ++t) {
          const v2f bv = w2p[pr * COUT + t * 16 + l16];  // ds_load_b64
          acc2[t] = __builtin_amdgcn_wmma_f32_16x16x4_f32(false, a, false, bv, (short)0, acc2[t], false, false);
        }
      }
      asm volatile("s_wait_dscnt 0" ::: "memory");       // reads done before next mt overwrites

      // ---- running max over valid rows (C-layout: lane L and L+16 share a column) ----
      #pragma unroll
      for (int t = 0; t < 4; ++t) {
        float m;
        if (mt == 0) {                                   // rows 0-15 all valid
          m = acc2[t][0];
          #pragma unroll
          for (int j = 1; j < 8; ++j) m = fmaxf(m, acc2[t][j]);
          m = fmaxf(m, __shfl_xor(m, 16));
        } else {                                         // only rows 0-3 (edges 16-19)
          m = acc2[t][0];
          #pragma unroll
          for (int j = 1; j < 4; ++j) m = fmaxf(m, acc2[t][j]);
          float o = __shfl_xor(m, 16);                   // bring lower-half value to upper
          m = half ? o : m;
        }
        rmax[t] = fmaxf(rmax[t], m);
      }
    }

    if (half == 0) {
      #pragma unroll
      for (int t = 0; t < 4; ++t) out[p * COUT + t * 16 + l16] = rmax[t];
    }
  }
}

extern "C" void kernel_launch(void* const* d_in, const int* in_sizes, int n_in,
                              void* d_out, int out_size, void* d_ws, size_t ws_size,
                              hipStream_t stream) {
  const float* x     = (const float*)d_in[0];
  // d_in[1] = batch indices (derivable, unused)
  const float* w1    = (const float*)d_in[2];
  const float* b1    = (const float*)d_in[3];
  const float* gamma = (const float*)d_in[4];
  const float* beta  = (const float*)d_in[5];
  const float* w2    = (const float*)d_in[6];
  const float* b2    = (const float*)d_in[7];
  float* out = (float*)d_out;

  char*  ws    = (char*)d_ws;
  int*   idxws = (int*)ws;                                // NEDGE ints (5.24 MB)
  float* stats = (float*)(ws + (size_t)NEDGE * sizeof(int));
  float* sbuf  = stats + 32;
  float* tbuf  = sbuf + COUT;

  k_zero<<<1, 32, 0, stream>>>(stats);
  k_knn<<<dim3(BB * 16), dim3(256), 0, stream>>>(x, idxws, stats);
  k_bnstats<<<1, 64, 0, stream>>>(stats, w1, b1, gamma, beta, sbuf, tbuf);
  k_edgeconv<<<dim3(NPTS / (WPB * PPW)), dim3(256), 0, stream>>>(
      x, idxws, w1, w2, b2, sbuf, tbuf, out);
}